// GraphLayer_4037269259012
// MI455X (gfx1250) — compile-verified
//
#include <hip/hip_runtime.h>
#include <math.h>

#define N_NODES 50000
#define E_EDGES 1600000
#define IN_F 128
#define HID_F 128
#define OUT_F 128
#define HEADS 4
#define ALPHA 0.2f
#define LN_EPS 1e-6f
#define INV_SCALE (1.0f / 22.627416997969522f)   // 1/sqrt(HID*HEADS)

typedef __attribute__((ext_vector_type(2))) float v2f;
typedef __attribute__((ext_vector_type(8))) float v8f;

// Guaranteed hardware fp32 atomic (no CAS loop). No-return form tracks via
// STOREcnt; the implicit wait-idle at s_endpgm guarantees completion before
// the next kernel in the stream observes agg/rowsum.
__device__ __forceinline__ void atomic_add_f32g(float* p, float v) {
    asm volatile("global_atomic_add_f32 %0, %1, off"
                 :: "v"(p), "v"(v)
                 : "memory");
}

// -------------------------------------------------------------------------
// Kernel 1: h[head] = data @ W[head]^T + b[head]
// One wave32 per 16x16 output tile; K=128 via 32x V_WMMA_F32_16X16X4_F32.
// -------------------------------------------------------------------------
__global__ void gemm_h_kernel(const float* __restrict__ data,
                              const float* __restrict__ Ww,
                              const float* __restrict__ Wb,
                              float* __restrict__ h) {
    const int lane = threadIdx.x;            // blockDim.x == 32
    const int m0   = blockIdx.x * 16;        // node tile
    const int j0   = blockIdx.y * 16;        // hidden-feature tile
    const int head = blockIdx.z;
    const int lh   = lane >> 4;              // 0 / 1  (upper lane half)
    const int ll   = lane & 15;

    const float* W = Ww + (size_t)head * HID_F * IN_F;  // [HID][IN] row-major
    const int mA = m0 + ll;                  // A-matrix row for this lane
    const int jB = j0 + ll;                  // B-matrix col for this lane

    v8f c = {};
    for (int kk = 0; kk < IN_F; kk += 4) {
        const int ka = kk + 2 * lh;          // K pair handled by this lane half
        v2f a, b;
        // A = data (16x4 f32 fragment): lane = m + 16*(k>>1), vgpr = k&1
        a.x = data[(size_t)mA * IN_F + ka];
        a.y = data[(size_t)mA * IN_F + ka + 1];
        // B[k][j] = W[j][k] (4x16 f32 fragment, mirrored layout)
        b.x = W[(size_t)jB * IN_F + ka];
        b.y = W[(size_t)jB * IN_F + ka + 1];
        c = __builtin_amdgcn_wmma_f32_16x16x4_f32(false, a, false, b,
                                                  (short)0, c, false, false);
    }
    const float bias = Wb[head * HID_F + jB];
    float* ho = h + (size_t)head * N_NODES * HID_F;
#pragma unroll
    for (int r = 0; r < 8; ++r) {
        const int m = m0 + r + 8 * lh;       // C/D layout: row = r + 8*(lane>>4)
        ho[(size_t)m * HID_F + jB] = c[r] + bias;
    }
}

// -------------------------------------------------------------------------
// Kernel 2: s_src = h . a[:128], s_dst = h . a[128:]  (wave per head-node)
// -------------------------------------------------------------------------
__global__ void score_kernel(const float* __restrict__ h,
                             const float* __restrict__ a,
                             float* __restrict__ s_src,
                             float* __restrict__ s_dst) {
    const int wave = (int)((blockIdx.x * blockDim.x + threadIdx.x) >> 5);
    const int lane = threadIdx.x & 31;
    if (wave >= N_NODES * HEADS) return;
    const int head = wave / N_NODES;
    const int node = wave - head * N_NODES;

    const float4 hv = *(const float4*)(h + ((size_t)head * N_NODES + node) * HID_F + lane * 4);
    const float4 as = *(const float4*)(a + head * 2 * HID_F + lane * 4);
    const float4 ad = *(const float4*)(a + head * 2 * HID_F + HID_F + lane * 4);
    float s1 = hv.x * as.x + hv.y * as.y + hv.z * as.z + hv.w * as.w;
    float s2 = hv.x * ad.x + hv.y * ad.y + hv.z * ad.z + hv.w * ad.w;
#pragma unroll
    for (int m = 16; m; m >>= 1) {
        s1 += __shfl_xor(s1, m, 32);
        s2 += __shfl_xor(s2, m, 32);
    }
    if (lane == 0) { s_src[wave] = s1; s_dst[wave] = s2; }
}

// -------------------------------------------------------------------------
// Kernel 3: edge scatter. One wave32 per edge; row/col forced into SGPRs so
// score reads become scalar loads and the h/agg accesses use saddr+voffset.
// Coalesced float4 gather of h[col]; hardware f32 atomics into agg/rowsum.
// -------------------------------------------------------------------------
__global__ void edge_scatter_kernel(const int* __restrict__ edge,
                                    const float* __restrict__ h,
                                    const float* __restrict__ s_src,
                                    const float* __restrict__ s_dst,
                                    float* __restrict__ agg,
                                    float* __restrict__ rowsum) {
    const int wave = (int)((blockIdx.x * blockDim.x + threadIdx.x) >> 5);
    const int lane = threadIdx.x & 31;
    if (wave >= E_EDGES) return;
    // wave index is uniform across the 32 lanes: pin indices in SGPRs
    const int row = __builtin_amdgcn_readfirstlane(edge[wave]);
    const int col = __builtin_amdgcn_readfirstlane(edge[E_EDGES + wave]);

#pragma unroll
    for (int head = 0; head < HEADS; ++head) {
        float e = s_src[head * N_NODES + row] + s_dst[head * N_NODES + col];
        e = (e > 0.0f ? e : ALPHA * e) * INV_SCALE;
        const float ee = __expf(e);
        if (lane == 0) atomic_add_f32g(&rowsum[head * N_NODES + row], ee);
        const float4 hv = *(const float4*)(h + ((size_t)head * N_NODES + col) * HID_F + lane * 4);
        float* ag = agg + ((size_t)head * N_NODES + row) * HID_F + lane * 4;
        atomic_add_f32g(ag + 0, ee * hv.x);
        atomic_add_f32g(ag + 1, ee * hv.y);
        atomic_add_f32g(ag + 2, ee * hv.z);
        atomic_add_f32g(ag + 3, ee * hv.w);
    }
}

// -------------------------------------------------------------------------
// Kernel 4: zero-row fixup + mean over heads + LayerNorm(ddof=1) + ReLU
// One wave32 per node; 4 features per lane.
// -------------------------------------------------------------------------
__global__ void finalize_kernel(const float* __restrict__ h,
                                const float* __restrict__ agg,
                                const float* __restrict__ rowsum,
                                const float* __restrict__ ln_g,
                                const float* __restrict__ ln_b,
                                float* __restrict__ xr) {
    const int node = (int)((blockIdx.x * blockDim.x + threadIdx.x) >> 5);
    const int lane = threadIdx.x & 31;
    if (node >= N_NODES) return;

    float hp[4] = {0.f, 0.f, 0.f, 0.f};
#pragma unroll
    for (int head = 0; head < HEADS; ++head) {
        float rs = rowsum[head * N_NODES + node];
        const size_t base = ((size_t)head * N_NODES + node) * HID_F + lane * 4;
        float4 av = *(const float4*)(agg + base);
        if (rs == 0.0f) {                    // isolated row: keep own h, denom 1
            av = *(const float4*)(h + base);
            rs = 1.0f;
        }
        const float inv = 1.0f / rs;
        hp[0] += av.x * inv; hp[1] += av.y * inv;
        hp[2] += av.z * inv; hp[3] += av.w * inv;
    }
#pragma unroll
    for (int i = 0; i < 4; ++i) hp[i] *= 0.25f;   // mean over heads

    float s = hp[0] + hp[1] + hp[2] + hp[3];
    float q = hp[0]*hp[0] + hp[1]*hp[1] + hp[2]*hp[2] + hp[3]*hp[3];
#pragma unroll
    for (int m = 16; m; m >>= 1) {
        s += __shfl_xor(s, m, 32);
        q += __shfl_xor(q, m, 32);
    }
    const float mean = s * (1.0f / 128.0f);
    const float var  = fmaxf((q - 128.0f * mean * mean) * (1.0f / 127.0f), 0.0f);
    const float inv  = 1.0f / (sqrtf(var) + LN_EPS);

    const float4 g  = *(const float4*)(ln_g + lane * 4);
    const float4 bb = *(const float4*)(ln_b + lane * 4);
    float4 o;
    o.x = fmaxf(g.x * (hp[0] - mean) * inv + bb.x, 0.0f);
    o.y = fmaxf(g.y * (hp[1] - mean) * inv + bb.y, 0.0f);
    o.z = fmaxf(g.z * (hp[2] - mean) * inv + bb.z, 0.0f);
    o.w = fmaxf(g.w * (hp[3] - mean) * inv + bb.w, 0.0f);
    *(float4*)(xr + (size_t)node * HID_F + lane * 4) = o;
}

// -------------------------------------------------------------------------
// Kernel 5: out = xr @ V_w^T + V_b  (same WMMA scheme as kernel 1)
// -------------------------------------------------------------------------
__global__ void gemm_out_kernel(const float* __restrict__ xr,
                                const float* __restrict__ Vw,
                                const float* __restrict__ Vb,
                                float* __restrict__ out) {
    const int lane = threadIdx.x;            // blockDim.x == 32
    const int m0 = blockIdx.x * 16;
    const int j0 = blockIdx.y * 16;
    const int lh = lane >> 4;
    const int ll = lane & 15;
    const int mA = m0 + ll;
    const int jB = j0 + ll;

    v8f c = {};
    for (int kk = 0; kk < HID_F; kk += 4) {
        const int ka = kk + 2 * lh;
        v2f a, b;
        a.x = xr[(size_t)mA * HID_F + ka];
        a.y = xr[(size_t)mA * HID_F + ka + 1];
        b.x = Vw[(size_t)jB * HID_F + ka];
        b.y = Vw[(size_t)jB * HID_F + ka + 1];
        c = __builtin_amdgcn_wmma_f32_16x16x4_f32(false, a, false, b,
                                                  (short)0, c, false, false);
    }
    const float bias = Vb[jB];
#pragma unroll
    for (int r = 0; r < 8; ++r) {
        const int m = m0 + r + 8 * lh;
        out[(size_t)m * OUT_F + jB] = c[r] + bias;
    }
}

extern "C" void kernel_launch(void* const* d_in, const int* in_sizes, int n_in,
                              void* d_out, int out_size, void* d_ws, size_t ws_size,
                              hipStream_t stream) {
    const int*   edge = (const int*)d_in[0];    // [2, E]
    const float* data = (const float*)d_in[1];  // [N, 128]
    const float* Ww   = (const float*)d_in[2];  // [4, 128, 128]
    const float* Wb   = (const float*)d_in[3];  // [4, 128]
    const float* a    = (const float*)d_in[4];  // [4, 256]
    const float* Vw   = (const float*)d_in[5];  // [128, 128]
    const float* Vb   = (const float*)d_in[6];  // [128]
    const float* lng  = (const float*)d_in[7];  // [128]
    const float* lnb  = (const float*)d_in[8];  // [128]
    float* out = (float*)d_out;

    // workspace carve-out (floats)
    float* ws     = (float*)d_ws;
    float* h      = ws;                                          // 25.6M
    float* agg    = h      + (size_t)HEADS * N_NODES * HID_F;    // 25.6M
    float* rowsum = agg    + (size_t)HEADS * N_NODES * HID_F;    // 200K
    float* s_src  = rowsum + (size_t)HEADS * N_NODES;            // 200K
    float* s_dst  = s_src  + (size_t)HEADS * N_NODES;            // 200K
    float* xr     = s_dst  + (size_t)HEADS * N_NODES;            // 6.4M

    // agg+rowsum are contiguous: zero both every call (atomics accumulate)
    hipMemsetAsync(agg, 0,
                   ((size_t)HEADS * N_NODES * HID_F + (size_t)HEADS * N_NODES) * sizeof(float),
                   stream);

    dim3 g1(N_NODES / 16, HID_F / 16, HEADS);          // 3125 x 8 x 4
    gemm_h_kernel<<<g1, 32, 0, stream>>>(data, Ww, Wb, h);

    score_kernel<<<(N_NODES * HEADS + 7) / 8, 256, 0, stream>>>(h, a, s_src, s_dst);

    edge_scatter_kernel<<<(E_EDGES + 7) / 8, 256, 0, stream>>>(edge, h, s_src, s_dst,
                                                               agg, rowsum);

    finalize_kernel<<<(N_NODES + 7) / 8, 256, 0, stream>>>(h, agg, rowsum, lng, lnb, xr);

    dim3 g5(N_NODES / 16, OUT_F / 16);                 // 3125 x 8
    gemm_out_kernel<<<g5, 32, 0, stream>>>(xr, Vw, Vb, out);
}